// MCRloss_69191923138588
// MI455X (gfx1250) — compile-verified
//
#include <hip/hip_runtime.h>
#include <math.h>

typedef __attribute__((ext_vector_type(2))) float v2f;
typedef __attribute__((ext_vector_type(8))) float v8f;

// Problem constants
#define NB    16      // batch (T)
#define NCH   32      // channels per branch
#define HIN   192
#define PO    24      // pooled H/W
#define HWP   576     // 24*24
#define MR    96      // m = 3*32
#define GSTRIDE (MR*MR)

// ---------------------------------------------------------------------------
// 1) 8x8 average pool: [16,32,192,192] -> [16,32,24,24]  (one fea per launch)
// ---------------------------------------------------------------------------
__global__ void pool_kernel(const float* __restrict__ in, float* __restrict__ out) {
    int gid = blockIdx.x * blockDim.x + threadIdx.x;     // 16*32*576 threads
    int ox = gid % PO;
    int oy = (gid / PO) % PO;
    int c  = (gid / HWP) % NCH;
    int b  = gid / (HWP * NCH);
    const float* p = in + (((size_t)(b * NCH + c)) * HIN + oy * 8) * HIN + ox * 8;
    float s = 0.f;
#pragma unroll
    for (int r = 0; r < 8; ++r) {
        const float4* q = (const float4*)(p + (size_t)r * HIN);
        float4 a = q[0];
        float4 d = q[1];
        s += a.x + a.y + a.z + a.w + d.x + d.y + d.z + d.w;
    }
    out[gid] = s * (1.0f / 64.0f);
}

// ---------------------------------------------------------------------------
// 2) ReflectionPad(1) + 3x3 conv (32->32, no bias) + LeakyReLU(0.2)
//    one block per (fea,b); 576 threads = one spatial position each.
//    Writes A[t=b][j=fea*32+oc][i=p]  (row-major 96x576 per t).
// ---------------------------------------------------------------------------
__device__ __forceinline__ int refl24(int i) {
    return i < 0 ? -i : (i > 23 ? 46 - i : i);
}

__global__ void conv_kernel(const float* __restrict__ pooled,
                            const float* __restrict__ W1,
                            const float* __restrict__ W2,
                            const float* __restrict__ W3,
                            float* __restrict__ A) {
    int blk = blockIdx.x;               // 0..47
    int f = blk / NB, b = blk % NB;
    const float* W = (f == 0) ? W1 : ((f == 1) ? W2 : W3);
    const float* in = pooled + (size_t)blk * (NCH * HWP);

    int p = threadIdx.x;                // 0..575
    int x = p % PO, y = p / PO;
    int yy[3], xx[3];
#pragma unroll
    for (int d = 0; d < 3; ++d) { yy[d] = refl24(y + d - 1); xx[d] = refl24(x + d - 1); }
    int nb[9];
#pragma unroll
    for (int dy = 0; dy < 3; ++dy)
#pragma unroll
        for (int dx = 0; dx < 3; ++dx) nb[dy * 3 + dx] = yy[dy] * PO + xx[dx];

    __shared__ float sIn[16 * HWP];     // 36 KB: 16 channels at a time
    float acc[NCH];
#pragma unroll
    for (int oc = 0; oc < NCH; ++oc) acc[oc] = 0.f;

    for (int phase = 0; phase < 2; ++phase) {
        for (int i = threadIdx.x; i < 16 * HWP; i += blockDim.x)
            sIn[i] = in[(size_t)phase * 16 * HWP + i];
        __syncthreads();
        for (int ic = 0; ic < 16; ++ic) {
            float v[9];
#pragma unroll
            for (int t = 0; t < 9; ++t) v[t] = sIn[ic * HWP + nb[t]];
            int icg = phase * 16 + ic;
#pragma unroll
            for (int oc = 0; oc < NCH; ++oc) {
                const float* w = W + ((size_t)oc * NCH + icg) * 9;  // uniform -> scalar loads
#pragma unroll
                for (int t = 0; t < 9; ++t) acc[oc] += v[t] * w[t];
            }
        }
        __syncthreads();
    }
#pragma unroll
    for (int oc = 0; oc < NCH; ++oc) {
        float v = acc[oc];
        v = v > 0.f ? v : 0.2f * v;                         // LeakyReLU(0.2)
        A[((size_t)b * MR + f * NCH + oc) * HWP + p] = v;
    }
}

// ---------------------------------------------------------------------------
// 3) Gram: G_t = A_t * A_t^T  (96x96, K=576) via V_WMMA_F32_16X16X4_F32.
//    One block per t, 8 waves, LDS-staged K-chunks of 96.
//    A-frag and B-frag are layout mirrors: lane&15 selects row (M) / col (N),
//    (lane>>4) selects K pair {0,1} vs {2,3}; each lane loads 2 K-consecutive f32.
// ---------------------------------------------------------------------------
__global__ void gram_kernel(const float* __restrict__ A, float* __restrict__ G) {
    const int S = 98;                    // LDS row stride (floats): 8B aligned pairs
    __shared__ float As[MR * S];         // 37,632 B
    int t = blockIdx.x;
    const float* At = A + (size_t)t * MR * HWP;
    int tid  = threadIdx.x;
    int w    = tid >> 5;                 // wave id 0..7
    int lane = tid & 31;
    int lm   = lane & 15;
    int half = lane >> 4;
    int ntile = (43 - w) >> 3;           // waves 0-3: 5 tiles, 4-7: 4 tiles

    v8f zero = {0.f, 0.f, 0.f, 0.f, 0.f, 0.f, 0.f, 0.f};
    v8f acc[5];
#pragma unroll
    for (int q = 0; q < 5; ++q) acc[q] = zero;

    for (int kc = 0; kc < HWP / MR; ++kc) {          // 6 chunks of K=96
        for (int idx = tid; idx < MR * MR; idx += 256) {
            int r = idx / MR, k = idx - r * MR;
            As[r * S + k] = At[(size_t)r * HWP + kc * MR + k];
        }
        __syncthreads();
        for (int q = 0; q < ntile; ++q) {
            int tile = w + (q << 3);                 // 0..35
            int tm = tile / 6, tn = tile % 6;
            int ra = tm * 16 + lm;                   // A-frag row
            int rb = tn * 16 + lm;                   // B-frag "row" (= G column block row)
            v8f c = acc[q];
            for (int kk = 0; kk < MR; kk += 4) {
                int ks = kk + 2 * half;
                v2f a  = *(const v2f*)&As[ra * S + ks];
                v2f bb = *(const v2f*)&As[rb * S + ks];
                c = __builtin_amdgcn_wmma_f32_16x16x4_f32(
                        false, a, false, bb, (short)0, c, false, false);
            }
            acc[q] = c;
        }
        __syncthreads();
    }
    // D layout: VGPR r -> M = r + 8*(lane>=16), N = lane&15
    for (int q = 0; q < ntile; ++q) {
        int tile = w + (q << 3);
        int tm = tile / 6, tn = tile % 6;
#pragma unroll
        for (int r = 0; r < 8; ++r) {
            int m = tm * 16 + r + 8 * half;
            int n = tn * 16 + lm;
            G[(size_t)t * GSTRIDE + m * MR + n] = acc[q][r];
        }
    }
}

// ---------------------------------------------------------------------------
// 4) logdet via Gaussian elimination (SPD, no pivoting), one block per t.
//    ld96[t] = logdet(I + 6*G_t);  ldc[t] = sum_c logdet(I + 18*G_t[block c])
// ---------------------------------------------------------------------------
__global__ void logdet_kernel(const float* __restrict__ G,
                              float* __restrict__ ld96, float* __restrict__ ldc) {
    __shared__ float M[MR * 97];         // 37,248 B
    __shared__ float Mb[3 * 32 * 33];    // 12,672 B
    int t = blockIdx.x;
    const float* Gt = G + (size_t)t * GSTRIDE;
    int tid = threadIdx.x;               // 128 threads

    for (int idx = tid; idx < MR * MR; idx += blockDim.x) {
        int i = idx / MR, j = idx - i * MR;
        M[i * 97 + j] = 6.0f * Gt[idx] + (i == j ? 1.0f : 0.0f);
    }
    for (int idx = tid; idx < 3 * 32 * 32; idx += blockDim.x) {
        int c = idx >> 10;
        int r = idx & 1023;
        int i = r >> 5, j = r & 31;
        Mb[c * (32 * 33) + i * 33 + j] =
            18.0f * Gt[(32 * c + i) * MR + 32 * c + j] + (i == j ? 1.0f : 0.0f);
    }
    __syncthreads();

    float lsum32 = 0.f;
    for (int c = 0; c < 3; ++c) {
        float* m = Mb + c * (32 * 33);
        for (int k = 0; k < 32; ++k) {
            float piv = m[k * 33 + k];
            if (tid == 0) lsum32 += logf(piv);
            for (int i = k + 1 + tid; i < 32; i += blockDim.x) {
                float fm = m[i * 33 + k] / piv;
                for (int j = k + 1; j < 32; ++j) m[i * 33 + j] -= fm * m[k * 33 + j];
            }
            __syncthreads();
        }
    }
    float lsum96 = 0.f;
    for (int k = 0; k < MR; ++k) {
        float piv = M[k * 97 + k];
        if (tid == 0) lsum96 += logf(piv);
        for (int i = k + 1 + tid; i < MR; i += blockDim.x) {
            float fm = M[i * 97 + k] / piv;
            for (int j = k + 1; j < MR; ++j) M[i * 97 + j] -= fm * M[k * 97 + j];
        }
        __syncthreads();
    }
    if (tid == 0) { ld96[t] = lsum96; ldc[t] = lsum32; }
}

// ---------------------------------------------------------------------------
// 5) loss = (sum_t ld96 - (1/3) sum_t ldc) / (2*T)   with 2*T = 32
// ---------------------------------------------------------------------------
__global__ void finalize_kernel(const float* __restrict__ ld96,
                                const float* __restrict__ ldc,
                                float* __restrict__ out) {
    if (threadIdx.x == 0 && blockIdx.x == 0) {
        float e = 0.f, cm = 0.f;
        for (int t = 0; t < NB; ++t) { e += ld96[t]; cm += ldc[t]; }
        out[0] = (e - cm * (1.0f / 3.0f)) * (1.0f / 32.0f);
    }
}

// ---------------------------------------------------------------------------
extern "C" void kernel_launch(void* const* d_in, const int* in_sizes, int n_in,
                              void* d_out, int out_size, void* d_ws, size_t ws_size,
                              hipStream_t stream) {
    (void)in_sizes; (void)n_in; (void)out_size; (void)ws_size;
    const float* feas[3] = {(const float*)d_in[0], (const float*)d_in[1], (const float*)d_in[2]};
    const float* W1 = (const float*)d_in[3];
    const float* W2 = (const float*)d_in[4];
    const float* W3 = (const float*)d_in[5];
    float* out = (float*)d_out;

    float* pooled = (float*)d_ws;                         // 3*16*32*576 f32 = 3.54 MB
    float* A      = pooled + (size_t)3 * NB * NCH * HWP;  // 16*96*576 f32  = 3.54 MB
    float* G      = A + (size_t)NB * MR * HWP;            // 16*96*96 f32   = 0.59 MB
    float* ld96   = G + (size_t)NB * GSTRIDE;             // 16 f32
    float* ldc    = ld96 + NB;                            // 16 f32

    const int poolThreads = NB * NCH * HWP;               // 294912 per fea
    for (int f = 0; f < 3; ++f)
        pool_kernel<<<poolThreads / 256, 256, 0, stream>>>(
            feas[f], pooled + (size_t)f * poolThreads);

    conv_kernel<<<3 * NB, HWP, 0, stream>>>(pooled, W1, W2, W3, A);
    gram_kernel<<<NB, 256, 0, stream>>>(A, G);
    logdet_kernel<<<NB, 128, 0, stream>>>(G, ld96, ldc);
    finalize_kernel<<<1, 32, 0, stream>>>(ld96, ldc, out);
}